// GraphAttentionV2Layer_60327110640265
// MI455X (gfx1250) — compile-verified
//
#include <hip/hip_runtime.h>

typedef float v2f __attribute__((ext_vector_type(2)));
typedef float v8f __attribute__((ext_vector_type(8)));

#define NN    1024
#define INF   128
#define NH    4
#define NF    16
#define HF    64   // NH*NF
#define NEG_SLOPE 0.2f

// ---------------------------------------------------------------------------
// Kernel 1: g = h @ W + b for both (W_l,b_l)->g_l and (W_r,b_r)->g_r
// One wave (32 lanes) per 16x16 output tile, fp32 WMMA 16x16x4.
// grid = (NN/16, HF/16, 2), block = 32
// ---------------------------------------------------------------------------
__global__ __launch_bounds__(32) void gat_gemm_gen(
    const float* __restrict__ hm,
    const float* __restrict__ Wl, const float* __restrict__ bl,
    const float* __restrict__ Wr, const float* __restrict__ br,
    float* __restrict__ gl, float* __restrict__ gr)
{
  const float* W;
  const float* b;
  float* g;
  if (blockIdx.z == 0) { W = Wl; b = bl; g = gl; }
  else                 { W = Wr; b = br; g = gr; }

  const int i0   = blockIdx.x * 16;
  const int c0   = blockIdx.y * 16;
  const int lane = threadIdx.x;        // 0..31
  const int r    = lane & 15;          // A: row / B: col / D: col
  const int kb   = (lane >> 4) << 1;   // k sub-offset: 0 (lanes 0-15) or 2 (16-31)

  v8f acc = {};
  const float* arow = hm + (size_t)(i0 + r) * INF;

#if __has_builtin(__builtin_amdgcn_wmma_f32_16x16x4_f32)
  #pragma unroll 4
  for (int k0 = 0; k0 < INF; k0 += 4) {
    v2f av, bv;
    av.x = arow[k0 + kb];
    av.y = arow[k0 + kb + 1];
    bv.x = W[(size_t)(k0 + kb)     * HF + c0 + r];
    bv.y = W[(size_t)(k0 + kb + 1) * HF + c0 + r];
    acc = __builtin_amdgcn_wmma_f32_16x16x4_f32(
        /*neg_a=*/false, av, /*neg_b=*/false, bv,
        /*c_mod=*/(short)0, acc, /*reuse_a=*/false, /*reuse_b=*/false);
  }
#else
  // scalar fallback (should not be taken on gfx1250 toolchains)
  for (int k0 = 0; k0 < INF; ++k0) {
    float a = arow[k0];
    for (int v = 0; v < 8; ++v)
      acc[v] += a * W[(size_t)k0 * HF + c0 + r];
  }
#endif

  const float bb = b[c0 + r];
  const int rb = (lane >> 4) * 8;      // D rows: v (lanes 0-15) or v+8 (16-31)
  #pragma unroll
  for (int v = 0; v < 8; ++v)
    g[(size_t)(i0 + rb + v) * HF + c0 + r] = acc[v] + bb;
}

// ---------------------------------------------------------------------------
// Kernel 2 (fused): per destination node i, compute e[i,j,h] =
//   sum_f lrelu(g_l[j,h,f] + g_r[i,h,f]) * w[f], mask by adj, softmax over j,
// write attention a to global exactly once. Row of e lives in LDS (16 KB).
// grid = NN, block = 256
// ---------------------------------------------------------------------------
__global__ __launch_bounds__(256) void gat_edge_softmax(
    const float* __restrict__ gl, const float* __restrict__ gr,
    const int* __restrict__ adj, const float* __restrict__ w,
    float* __restrict__ a)
{
  __shared__ float s_e[NN * NH];     // 16 KB: e row for all (j,h)
  __shared__ float s_gr[HF];
  __shared__ float s_w[NF];
  __shared__ float s_red[256];

  const int i   = blockIdx.x;
  const int tid = threadIdx.x;

  if (tid < HF) s_gr[tid] = gr[(size_t)i * HF + tid];
  if (tid < NF) s_w[tid]  = w[tid];
  __syncthreads();

  // ---- edge scores + mask ----
  for (int j = tid; j < NN; j += 256) {
    const float* glr = gl + (size_t)j * HF;
    const int m = adj[(size_t)i * NN + j];
    #pragma unroll
    for (int h = 0; h < NH; ++h) {
      float acc = 0.0f;
      #pragma unroll
      for (int f = 0; f < NF; ++f) {
        float s = glr[h * NF + f] + s_gr[h * NF + f];
        float l = fmaxf(s, NEG_SLOPE * s);   // leaky_relu, slope<1
        acc = __builtin_fmaf(l, s_w[f], acc);
      }
      s_e[j * NH + h] = (m == 0) ? -1e20f : acc;
    }
  }
  __syncthreads();

  // ---- per-head softmax over j ----
  for (int h = 0; h < NH; ++h) {
    float lm = -3.0e38f;
    for (int j = tid; j < NN; j += 256) lm = fmaxf(lm, s_e[j * NH + h]);
    s_red[tid] = lm; __syncthreads();
    for (int s = 128; s > 0; s >>= 1) {
      if (tid < s) s_red[tid] = fmaxf(s_red[tid], s_red[tid + s]);
      __syncthreads();
    }
    const float mx = s_red[0];
    __syncthreads();

    float ls = 0.0f;
    for (int j = tid; j < NN; j += 256) {
      float ev = __expf(s_e[j * NH + h] - mx);
      s_e[j * NH + h] = ev;
      ls += ev;
    }
    s_red[tid] = ls; __syncthreads();
    for (int s = 128; s > 0; s >>= 1) {
      if (tid < s) s_red[tid] += s_red[tid + s];
      __syncthreads();
    }
    const float inv = 1.0f / s_red[0];
    __syncthreads();

    for (int j = tid; j < NN; j += 256)
      a[((size_t)i * NN + j) * NH + h] = s_e[j * NH + h] * inv;
  }
}

// ---------------------------------------------------------------------------
// Kernel 3: attn_res[i,h,f] = sum_j a[i,j,h] * g_r[j,h,f]
// Per head: (1024x1024) @ (1024x16) via fp32 WMMA 16x16x4, one wave per
// 16-row tile of i.  grid = (NN/16, NH), block = 32
// ---------------------------------------------------------------------------
__global__ __launch_bounds__(32) void gat_attn_out(
    const float* __restrict__ a, const float* __restrict__ gr,
    float* __restrict__ out)
{
  const int i0   = blockIdx.x * 16;
  const int h    = blockIdx.y;
  const int lane = threadIdx.x;
  const int r    = lane & 15;
  const int kb   = (lane >> 4) << 1;

  v8f acc = {};
  const float* arow = a + ((size_t)(i0 + r) * NN) * NH + h;   // stride NH over j
  const float* bcol = gr + h * NF + r;                        // stride HF over j

#if __has_builtin(__builtin_amdgcn_wmma_f32_16x16x4_f32)
  for (int j0 = 0; j0 < NN; j0 += 4) {
    v2f av, bv;
    av.x = arow[(size_t)(j0 + kb)     * NH];
    av.y = arow[(size_t)(j0 + kb + 1) * NH];
    bv.x = bcol[(size_t)(j0 + kb)     * HF];
    bv.y = bcol[(size_t)(j0 + kb + 1) * HF];
    acc = __builtin_amdgcn_wmma_f32_16x16x4_f32(
        false, av, false, bv, (short)0, acc, false, false);
  }
#else
  for (int j0 = 0; j0 < NN; ++j0) {
    float bvv = bcol[(size_t)j0 * HF];
    for (int v = 0; v < 8; ++v)
      acc[v] += a[((size_t)(i0 + v) * NN + j0) * NH + h] * bvv;
  }
#endif

  const int rb = (lane >> 4) * 8;
  #pragma unroll
  for (int v = 0; v < 8; ++v)
    out[(size_t)(i0 + rb + v) * HF + h * NF + r] = acc[v];
}

// ---------------------------------------------------------------------------
extern "C" void kernel_launch(void* const* d_in, const int* in_sizes, int n_in,
                              void* d_out, int out_size, void* d_ws, size_t ws_size,
                              hipStream_t stream) {
  (void)in_sizes; (void)n_in; (void)out_size; (void)ws_size;

  const float* h    = (const float*)d_in[0];   // (1,1024,128)
  const int*   adj  = (const int*)  d_in[1];   // (1,1024,1024)
  const float* Wl   = (const float*)d_in[2];   // (128,64)
  const float* bl   = (const float*)d_in[3];   // (64,)
  const float* Wr   = (const float*)d_in[4];   // (128,64)
  const float* br   = (const float*)d_in[5];   // (64,)
  const float* attw = (const float*)d_in[6];   // (16,)

  float* a_out   = (float*)d_out;                       // (1,1024,1024,4)
  float* res_out = a_out + (size_t)NN * NN * NH;        // (1,1024,64)

  float* gl = (float*)d_ws;            // 1024*64 floats
  float* gr = gl + (size_t)NN * HF;    // 1024*64 floats

  gat_gemm_gen<<<dim3(NN / 16, HF / 16, 2), 32, 0, stream>>>(
      h, Wl, bl, Wr, br, gl, gr);

  gat_edge_softmax<<<dim3(NN), 256, 0, stream>>>(gl, gr, adj, attw, a_out);

  gat_attn_out<<<dim3(NN / 16, NH), 32, 0, stream>>>(a_out, gr, res_out);
}